// KernelPool_57724360458781
// MI455X (gfx1250) — compile-verified
//
#include <hip/hip_runtime.h>

typedef __attribute__((ext_vector_type(2))) float v2f;
typedef __attribute__((ext_vector_type(8))) float v8f;

#define NPTS    4096
#define KOUT    512
#define WDIM    4
#define DDIM    2
#define NSLICES 64
#define ALPHA_R 1e-6f
#define NBLK    32           // KOUT / 16 panels

// ---------------------------------------------------------------------------
// Kernel 1: per-slice top-512 by weight-norm via bitonic sort of packed keys.
// key = (~bits(||w||^2) << 32) | index  -> ascending sort == descending norm,
// ties broken by smaller index (matches jax.lax.top_k).
// ---------------------------------------------------------------------------
__global__ __launch_bounds__(1024) void KP_topk(const float* __restrict__ wts,
                                                unsigned int* __restrict__ idx_out) {
    __shared__ unsigned long long keys[NPTS];
    const int s = blockIdx.x;
    const int tid = threadIdx.x;
    const float4* wb = (const float4*)(wts + (size_t)s * NPTS * WDIM);
    for (int e = tid; e < NPTS; e += 1024) {
        float4 w = wb[e];
        float n2 = w.x * w.x + w.y * w.y + w.z * w.z + w.w * w.w;
        unsigned int fb = __float_as_uint(n2);  // n2 >= 0: bits are order-preserving
        keys[e] = ((unsigned long long)(~fb) << 32) | (unsigned int)e;
    }
    for (int k = 2; k <= NPTS; k <<= 1) {
        for (int j = k >> 1; j > 0; j >>= 1) {
            __syncthreads();
            for (int i = tid; i < NPTS; i += 1024) {
                int ixj = i ^ j;
                if (ixj > i) {
                    unsigned long long a = keys[i], b = keys[ixj];
                    bool up = ((i & k) == 0);
                    if ((a > b) == up) { keys[i] = b; keys[ixj] = a; }
                }
            }
        }
    }
    __syncthreads();
    if (tid < KOUT) idx_out[s * KOUT + tid] = (unsigned int)(keys[tid] & 0xffffffffULL);
}

// ---------------------------------------------------------------------------
// Kernel 2: gather selected positions -> out_pos segment of d_out.
// ---------------------------------------------------------------------------
__global__ __launch_bounds__(KOUT) void KP_gather(const float* __restrict__ pos,
                                                  const unsigned int* __restrict__ idx,
                                                  float* __restrict__ out_pos) {
    const int s = blockIdx.x;
    const int k = threadIdx.x;
    unsigned int id = idx[s * KOUT + k];
    const float2* pp = (const float2*)(pos + (size_t)s * NPTS * DDIM);
    float2* op = (float2*)(out_pos + (size_t)s * KOUT * DDIM);
    op[k] = pp[id];
}

// Rank-4 distance-embedding operands (ISA 32-bit A 16x4 / B 4x16 lane layouts).
__device__ __forceinline__ v2f make_a(int half, float ox, float oy) {
    v2f a;
    a.x = half ? (ox * ox + oy * oy) : ox;
    a.y = half ? 1.0f : oy;
    return a;
}
__device__ __forceinline__ v2f make_b(int half, float px, float py) {
    v2f b;
    b.x = half ? 1.0f : (-2.0f * px);
    b.y = half ? (px * px + py * py) : (-2.0f * py);
    return b;
}

// ---------------------------------------------------------------------------
// Kernel 3: samples = exp(-d2/2) @ weights, fused.  One workgroup (4 waves)
// per (slice, 16-row tile); one v_wmma_f32_16x16x4_f32 per 16-column tile.
// ---------------------------------------------------------------------------
__global__ __launch_bounds__(128) void KP_samples(const float* __restrict__ pos,
                                                  const float* __restrict__ wts,
                                                  const float* __restrict__ out_pos,
                                                  float* __restrict__ samples) {
    const int s = blockIdx.y, rt = blockIdx.x;
    const int wave = threadIdx.x >> 5, lane = threadIdx.x & 31;
    const int half = lane >> 4, r = lane & 15;
    const float* pp = pos + (size_t)s * NPTS * DDIM;
    const float* op = out_pos + (size_t)s * KOUT * DDIM;
    const float4* wb = (const float4*)(wts + (size_t)s * NPTS * WDIM);

    float ox = op[(rt * 16 + r) * 2 + 0];
    float oy = op[(rt * 16 + r) * 2 + 1];
    v2f a = make_a(half, ox, oy);

    float acc[8][4];
#pragma unroll
    for (int q = 0; q < 8; ++q) { acc[q][0] = 0.f; acc[q][1] = 0.f; acc[q][2] = 0.f; acc[q][3] = 0.f; }

    for (int ct = wave; ct < NPTS / 16; ct += 4) {
        int n = ct * 16 + r;
        __builtin_prefetch((const void*)(wb + n + 64), 0, 1);   // next tile for this wave
        float px = pp[n * 2 + 0], py = pp[n * 2 + 1];
        v2f b = make_b(half, px, py);
        v8f c = {0.f, 0.f, 0.f, 0.f, 0.f, 0.f, 0.f, 0.f};
        c = __builtin_amdgcn_wmma_f32_16x16x4_f32(false, a, false, b, (short)0, c, false, false);
        float4 w = wb[n];
#pragma unroll
        for (int q = 0; q < 8; ++q) {
            float kc = __expf(-0.5f * fmaxf(c[q], 0.0f));
            acc[q][0] += kc * w.x;
            acc[q][1] += kc * w.y;
            acc[q][2] += kc * w.z;
            acc[q][3] += kc * w.w;
        }
    }

    __shared__ float ssum[16][4];
    if (threadIdx.x < 64) ((float*)ssum)[threadIdx.x] = 0.0f;
    __syncthreads();
#pragma unroll
    for (int q = 0; q < 8; ++q) {
        int M = q + half * 8;  // D-matrix VGPR->row mapping
        atomicAdd(&ssum[M][0], acc[q][0]);
        atomicAdd(&ssum[M][1], acc[q][1]);
        atomicAdd(&ssum[M][2], acc[q][2]);
        atomicAdd(&ssum[M][3], acc[q][3]);
    }
    __syncthreads();
    if (threadIdx.x < 64) {
        int M = threadIdx.x >> 2, j = threadIdx.x & 3;
        samples[(size_t)s * KOUT * WDIM + (rt * 16 + M) * WDIM + j] = ssum[M][j];
    }
}

// ---------------------------------------------------------------------------
// Kernel 4: Gram matrix K + alpha*I (512x512 per slice) via the same WMMA trick.
// ---------------------------------------------------------------------------
__global__ __launch_bounds__(128) void KP_gram(const float* __restrict__ out_pos,
                                               float* __restrict__ Kws) {
    const int s = blockIdx.y, rt = blockIdx.x;
    const int wave = threadIdx.x >> 5, lane = threadIdx.x & 31;
    const int half = lane >> 4, r = lane & 15;
    const float* op = out_pos + (size_t)s * KOUT * DDIM;
    float* Km = Kws + (size_t)s * KOUT * KOUT;

    float ox = op[(rt * 16 + r) * 2 + 0];
    float oy = op[(rt * 16 + r) * 2 + 1];
    v2f a = make_a(half, ox, oy);

    for (int ct = wave; ct < NBLK; ct += 4) {
        int n = ct * 16 + r;
        float px = op[n * 2 + 0], py = op[n * 2 + 1];
        v2f b = make_b(half, px, py);
        v8f c = {0.f, 0.f, 0.f, 0.f, 0.f, 0.f, 0.f, 0.f};
        c = __builtin_amdgcn_wmma_f32_16x16x4_f32(false, a, false, b, (short)0, c, false, false);
#pragma unroll
        for (int q = 0; q < 8; ++q) {
            int M = rt * 16 + q + half * 8;
            int Nc = ct * 16 + r;
            float kc = __expf(-0.5f * fmaxf(c[q], 0.0f));
            kc += (M == Nc) ? ALPHA_R : 0.0f;
            Km[(size_t)M * KOUT + Nc] = kc;
        }
    }
}

// ---------------------------------------------------------------------------
// Kernel 5: per-slice SPD solve.  Blocked right-looking Cholesky (NB=16):
//   1) factor 16x16 diagonal block (LDS),
//   2) row-parallel triangular solve of the panel L21 (cached in LDS),
//   3) trailing SYRK update A22 -= L21*L21^T with v_wmma_f32_16x16x4_f32
//      (4 chained K=4 WMMAs per 16x16 lower-triangle tile, operands from LDS,
//       A operand negated at load since f32 WMMA has no A/B NEG).
// Then column-sweep forward/backward substitution on the 4 RHS (float4 LDS).
// ---------------------------------------------------------------------------
__global__ __launch_bounds__(256) void KP_solve(float* __restrict__ Kws,
                                                const float* __restrict__ samples,
                                                float* __restrict__ out_w) {
    const int s = blockIdx.x, tid = threadIdx.x;
    const int wave = tid >> 5, lane = tid & 31;
    const int half = lane >> 4, r = lane & 15;
    float* A = Kws + (size_t)s * KOUT * KOUT;

    __shared__ float4 sb[KOUT];                 // 8 KB : RHS
    __shared__ float  sL11[16 * 16];            // 1 KB : diagonal block
    __shared__ float  sP[(KOUT - 16) * 16];     // 31 KB: current panel L21

    const float4* smp = (const float4*)(samples + (size_t)s * KOUT * WDIM);
    for (int i = tid; i < KOUT; i += 256) sb[i] = smp[i];

    // ---------------- blocked Cholesky ----------------
    for (int p = 0; p < NBLK; ++p) {
        const int c = p * 16;
        const int R16 = KOUT - c - 16;          // rows below the panel

        // load diagonal block
        {
            int row = tid >> 4, col = tid & 15;
            sL11[row * 16 + col] = A[(size_t)(c + row) * KOUT + c + col];
        }
        __syncthreads();

        // factor 16x16 (serial, trivial work)
        if (tid == 0) {
            for (int kk = 0; kk < 16; ++kk) {
                float d = sqrtf(fmaxf(sL11[kk * 16 + kk], 1e-30f));
                sL11[kk * 16 + kk] = d;
                float inv = 1.0f / d;
                for (int i = kk + 1; i < 16; ++i) sL11[i * 16 + kk] *= inv;
                for (int i = kk + 1; i < 16; ++i) {
                    float lik = sL11[i * 16 + kk];
                    for (int m = kk + 1; m <= i; ++m) sL11[i * 16 + m] -= lik * sL11[m * 16 + kk];
                }
            }
        }
        __syncthreads();

        // write factored diagonal block back; panel solve rows in parallel
        {
            int row = tid >> 4, col = tid & 15;
            A[(size_t)(c + row) * KOUT + c + col] = sL11[row * 16 + col];
        }
        for (int ii = tid; ii < R16; ii += 256) {
            int gi = c + 16 + ii;
            float a[16];
#pragma unroll
            for (int m = 0; m < 16; ++m) a[m] = A[(size_t)gi * KOUT + c + m];
#pragma unroll
            for (int kk = 0; kk < 16; ++kk) {
                float x = a[kk];
                for (int m = 0; m < kk; ++m) x -= sL11[kk * 16 + m] * a[m];
                a[kk] = x / sL11[kk * 16 + kk];
            }
#pragma unroll
            for (int m = 0; m < 16; ++m) {
                A[(size_t)gi * KOUT + c + m] = a[m];
                sP[ii * 16 + m] = a[m];
            }
        }
        __threadfence();
        __syncthreads();

        // trailing update: lower-triangle 16x16 tiles, distributed over 8 waves
        const int Rb = R16 >> 4;                // remaining block-rows
        const int T  = (Rb * (Rb + 1)) >> 1;    // lower-triangle tile count
        for (int t = wave; t < T; t += 8) {
            // decode triangular index t -> (bi, bk), bk <= bi
            int bi = (int)((sqrtf(8.0f * (float)t + 1.0f) - 1.0f) * 0.5f);
            while ((bi + 1) * (bi + 2) / 2 <= t) ++bi;
            while (bi * (bi + 1) / 2 > t) --bi;
            int bk = t - bi * (bi + 1) / 2;

            const int gi0 = c + 16 + bi * 16;
            const int gk0 = c + 16 + bk * 16;
            const int pi = bi * 16, pk = bk * 16;

            v8f cc;
#pragma unroll
            for (int q = 0; q < 8; ++q)
                cc[q] = A[(size_t)(gi0 + q + 8 * half) * KOUT + gk0 + r];

#pragma unroll
            for (int kk0 = 0; kk0 < 16; kk0 += 4) {
                int ksel = kk0 + 2 * half;
                v2f a, b;
                a.x = -sP[(pi + r) * 16 + ksel];
                a.y = -sP[(pi + r) * 16 + ksel + 1];
                b.x =  sP[(pk + r) * 16 + ksel];
                b.y =  sP[(pk + r) * 16 + ksel + 1];
                cc = __builtin_amdgcn_wmma_f32_16x16x4_f32(false, a, false, b, (short)0, cc, false, false);
            }
#pragma unroll
            for (int q = 0; q < 8; ++q)
                A[(size_t)(gi0 + q + 8 * half) * KOUT + gk0 + r] = cc[q];
        }
        __threadfence();
        __syncthreads();
    }

    // ---------------- forward substitution: L y = b ----------------
    for (int j = 0; j < KOUT; ++j) {
        if (tid == 0) {
            float inv = 1.0f / A[(size_t)j * KOUT + j];
            sb[j].x *= inv; sb[j].y *= inv; sb[j].z *= inv; sb[j].w *= inv;
        }
        __syncthreads();
        float4 xj = sb[j];
        for (int i = j + 1 + tid; i < KOUT; i += 256) {
            float lij = A[(size_t)i * KOUT + j];
            float4 v = sb[i];
            v.x -= lij * xj.x; v.y -= lij * xj.y; v.z -= lij * xj.z; v.w -= lij * xj.w;
            sb[i] = v;
        }
        __syncthreads();
    }

    // ---------------- backward substitution: L^T x = y ----------------
    for (int j = KOUT - 1; j >= 0; --j) {
        if (tid == 0) {
            float inv = 1.0f / A[(size_t)j * KOUT + j];
            sb[j].x *= inv; sb[j].y *= inv; sb[j].z *= inv; sb[j].w *= inv;
        }
        __syncthreads();
        float4 xj = sb[j];
        for (int i = tid; i < j; i += 256) {
            float lji = A[(size_t)j * KOUT + i];   // row j of L: contiguous
            float4 v = sb[i];
            v.x -= lji * xj.x; v.y -= lji * xj.y; v.z -= lji * xj.z; v.w -= lji * xj.w;
            sb[i] = v;
        }
        __syncthreads();
    }

    float4* ow = (float4*)(out_w + (size_t)s * KOUT * WDIM);
    for (int i = tid; i < KOUT; i += 256) ow[i] = sb[i];
}

// ---------------------------------------------------------------------------
// Host launcher.
// d_in[0] = input_positions (2,32,4096,2) f32 ; d_in[1] = input_weights (2,32,4096,4) f32
// d_out   = out_pos (65536 f32) || out_weights (131072 f32)
// d_ws    = [topk idx u32: 128KB][samples f32: 512KB][Gram f32: 64MB]
// ---------------------------------------------------------------------------
extern "C" void kernel_launch(void* const* d_in, const int* in_sizes, int n_in,
                              void* d_out, int out_size, void* d_ws, size_t ws_size,
                              hipStream_t stream) {
    (void)in_sizes; (void)n_in; (void)out_size; (void)ws_size;
    const float* d_pos = (const float*)d_in[0];
    const float* d_wts = (const float*)d_in[1];
    float* out_pos = (float*)d_out;
    float* out_w   = (float*)d_out + (size_t)NSLICES * KOUT * DDIM;

    unsigned char* ws = (unsigned char*)d_ws;
    unsigned int* idx = (unsigned int*)ws;
    float* samples = (float*)(ws + (size_t)NSLICES * KOUT * sizeof(unsigned int));
    float* Kws = (float*)(ws + (size_t)NSLICES * KOUT * sizeof(unsigned int)
                             + (size_t)NSLICES * KOUT * WDIM * sizeof(float));

    KP_topk<<<NSLICES, 1024, 0, stream>>>(d_wts, idx);
    KP_gather<<<NSLICES, KOUT, 0, stream>>>(d_pos, idx, out_pos);
    KP_samples<<<dim3(KOUT / 16, NSLICES), 128, 0, stream>>>(d_pos, d_wts, out_pos, samples);
    KP_gram<<<dim3(KOUT / 16, NSLICES), 128, 0, stream>>>(out_pos, Kws);
    KP_solve<<<NSLICES, 256, 0, stream>>>(Kws, samples, out_w);
}